// IterNormRotation_79594333929904
// MI455X (gfx1250) — compile-verified
//
#include <hip/hip_runtime.h>
#include <hip/hip_bf16.h>
#include <math.h>

typedef __attribute__((ext_vector_type(2))) float v2f;
typedef __attribute__((ext_vector_type(8))) float v8f;

#define C_DIM   256
#define B_DIM   128
#define HW      784                 // 28*28, 784 % 16 == 0
#define M_TOT   (B_DIM * HW)        // 100352
#define RIDGE_EPS 1e-5f
#define T_ITERS 10

#define WMMA_F32(a, b, c) \
    __builtin_amdgcn_wmma_f32_16x16x4_f32(false, (a), false, (b), (short)0, (c), false, false)

// ---------------------------------------------------------------- mean ------
__global__ void mean_kernel(const float* __restrict__ X, float* __restrict__ mean) {
    __shared__ float red[256];
    const int c = blockIdx.x;
    const int tid = threadIdx.x;
    float acc = 0.f;
    for (int b = 0; b < B_DIM; ++b) {
        const float* p = X + ((size_t)(b * C_DIM + c)) * HW;
        for (int i = tid; i < HW; i += 256) acc += p[i];
    }
    red[tid] = acc;
    __syncthreads();
    for (int s = 128; s > 0; s >>= 1) {
        if (tid < s) red[tid] += red[tid + s];
        __syncthreads();
    }
    if (tid == 0) mean[c] = red[0] * (1.0f / (float)M_TOT);
}

// ------------------------------------------------------- Sigma = eps*I ------
__global__ void sigma_init_kernel(float* __restrict__ Sigma) {
    const int i = blockIdx.x * blockDim.x + threadIdx.x;
    if (i < C_DIM * C_DIM) {
        const int r = i / C_DIM, cc = i - r * C_DIM;
        Sigma[i] = (r == cc) ? RIDGE_EPS : 0.0f;
    }
}

// -------------------------------------------- Sigma += xc @ xc^T / m --------
// One wave computes a 32x32 output tile (2x2 WMMA accumulators) for one image.
// Inner loop walks hw contiguously: per-lane A/B loads are adjacent pairs
// (-> global_load_b64) and there is no integer division in the hot loop.
__global__ void __launch_bounds__(32)
sigma_kernel(const float* __restrict__ X, const float* __restrict__ mean,
             float* __restrict__ Sigma) {
    const int tib = blockIdx.x * 2;      // 16-row tile base (0..14)
    const int tjb = blockIdx.y * 2;      // 16-col tile base
    const int img = blockIdx.z;          // batch image
    const int lane = threadIdx.x;
    const int rr = lane & 15;            // M/N index within a 16x16 tile
    const int g  = lane >> 4;            // k sub-group: k = kk + g*2 + {0,1}

    const int ca0 = tib * 16 + rr, ca1 = ca0 + 16;
    const int cb0 = tjb * 16 + rr, cb1 = cb0 + 16;
    const float mA0 = mean[ca0], mA1 = mean[ca1];
    const float mB0 = mean[cb0], mB1 = mean[cb1];

    const float* base = X + (size_t)img * C_DIM * HW;
    const float* pa0 = base + (size_t)ca0 * HW;
    const float* pa1 = base + (size_t)ca1 * HW;
    const float* pb0 = base + (size_t)cb0 * HW;
    const float* pb1 = base + (size_t)cb1 * HW;

    v8f acc00 = {}, acc01 = {}, acc10 = {}, acc11 = {};
    for (int hw = 0; hw < HW; hw += 4) {
        const int h0 = hw + g * 2;
        v2f a0, a1, b0, b1;
        a0.x = pa0[h0] - mA0;  a0.y = pa0[h0 + 1] - mA0;   // contiguous pair
        a1.x = pa1[h0] - mA1;  a1.y = pa1[h0 + 1] - mA1;
        b0.x = pb0[h0] - mB0;  b0.y = pb0[h0 + 1] - mB0;
        b1.x = pb1[h0] - mB1;  b1.y = pb1[h0 + 1] - mB1;
        acc00 = WMMA_F32(a0, b0, acc00);
        acc01 = WMMA_F32(a0, b1, acc01);
        acc10 = WMMA_F32(a1, b0, acc10);
        acc11 = WMMA_F32(a1, b1, acc11);
    }

    const float invm = 1.0f / (float)M_TOT;
    const int rb0 = tib * 16 + g * 8;    // C/D layout: VGPR r -> M=r (+8 lanes 16..31)
    const int rb1 = rb0 + 16;
    const int c0 = tjb * 16 + rr;
    const int c1 = c0 + 16;
#pragma unroll
    for (int r = 0; r < 8; ++r) {
        atomicAdd(&Sigma[(rb0 + r) * C_DIM + c0], acc00[r] * invm);
        atomicAdd(&Sigma[(rb0 + r) * C_DIM + c1], acc01[r] * invm);
        atomicAdd(&Sigma[(rb1 + r) * C_DIM + c0], acc10[r] * invm);
        atomicAdd(&Sigma[(rb1 + r) * C_DIM + c1], acc11[r] * invm);
    }
}

// ------------------------------------------------ rTr = 1/trace(Sigma) ------
__global__ void trace_kernel(const float* __restrict__ Sigma, float* __restrict__ scalars) {
    __shared__ float red[256];
    const int tid = threadIdx.x;
    red[tid] = Sigma[tid * C_DIM + tid];
    __syncthreads();
    for (int s = 128; s > 0; s >>= 1) {
        if (tid < s) red[tid] += red[tid + s];
        __syncthreads();
    }
    if (tid == 0) {
        const float rTr = 1.0f / red[0];
        scalars[0] = rTr;
        scalars[1] = sqrtf(rTr);
    }
}

// ------------------------------- Sigma_N = Sigma * rTr ; P = I --------------
__global__ void scale_init_kernel(const float* __restrict__ Sigma,
                                  const float* __restrict__ scalars,
                                  float* __restrict__ SigmaN,
                                  float* __restrict__ P) {
    const int i = blockIdx.x * blockDim.x + threadIdx.x;
    if (i < C_DIM * C_DIM) {
        SigmaN[i] = Sigma[i] * scalars[0];
        const int r = i / C_DIM, cc = i - r * C_DIM;
        P[i] = (r == cc) ? 1.0f : 0.0f;
    }
}

// --------------------------------------- Cout = A @ B (256x256x256) ---------
__global__ void __launch_bounds__(32)
matmul256_kernel(const float* __restrict__ A, const float* __restrict__ B,
                 float* __restrict__ Cout) {
    const int ti = blockIdx.x, tj = blockIdx.y;
    const int lane = threadIdx.x;
    const int rr = lane & 15;
    const int g  = lane >> 4;
    const float* arow = A + (ti * 16 + rr) * C_DIM;
    v8f acc = {};
    for (int k = 0; k < C_DIM; k += 4) {
        const int k0 = k + g * 2;
        v2f a, b;
        a.x = arow[k0];                       // contiguous pair -> b64
        a.y = arow[k0 + 1];
        b.x = B[k0 * C_DIM + tj * 16 + rr];
        b.y = B[(k0 + 1) * C_DIM + tj * 16 + rr];
        acc = WMMA_F32(a, b, acc);
    }
    const int col = tj * 16 + rr;
    const int rowbase = ti * 16 + g * 8;
#pragma unroll
    for (int r = 0; r < 8; ++r)
        Cout[(rowbase + r) * C_DIM + col] = acc[r];
}

// --------------------------- P = 1.5*P - 0.5*(T2 @ Sigma_N) -----------------
__global__ void __launch_bounds__(32)
newton_update_kernel(const float* __restrict__ T2, const float* __restrict__ SigmaN,
                     float* __restrict__ P) {
    const int ti = blockIdx.x, tj = blockIdx.y;
    const int lane = threadIdx.x;
    const int rr = lane & 15;
    const int g  = lane >> 4;
    const float* arow = T2 + (ti * 16 + rr) * C_DIM;
    v8f acc = {};
    for (int k = 0; k < C_DIM; k += 4) {
        const int k0 = k + g * 2;
        v2f a, b;
        a.x = arow[k0];
        a.y = arow[k0 + 1];
        b.x = SigmaN[k0 * C_DIM + tj * 16 + rr];
        b.y = SigmaN[(k0 + 1) * C_DIM + tj * 16 + rr];
        acc = WMMA_F32(a, b, acc);
    }
    const int col = tj * 16 + rr;
    const int rowbase = ti * 16 + g * 8;
#pragma unroll
    for (int r = 0; r < 8; ++r) {
        const int idx = (rowbase + r) * C_DIM + col;  // each element owned by one lane
        P[idx] = 1.5f * P[idx] - 0.5f * acc[r];
    }
}

// ----------------- out = (sqrt(rTr) * P) @ xc, written as [B,C,H,W] ---------
// One wave computes 4 row-tiles (64 channels) x 16 columns: the expensive
// strided B column loads are done once per k-step and reused for 4 WMMAs,
// and X is streamed only 4x through L2 instead of 16x.
__global__ void __launch_bounds__(32)
whiten_kernel(const float* __restrict__ X, const float* __restrict__ mean,
              const float* __restrict__ P, const float* __restrict__ scalars,
              float* __restrict__ out) {
    const int ti0 = blockIdx.x * 4;     // 16-row tile base: 0,4,8,12
    const int tj  = blockIdx.y;         // spatial-column tile 0..6271
    const int lane = threadIdx.x;
    const int rr = lane & 15;
    const int g  = lane >> 4;
    const int col = tj * 16 + rr;       // global column in [0, m)
    const int bb  = col / HW;           // tile never crosses an image (784 % 16 == 0)
    const int hw  = col - bb * HW;

    const float* arow = P + ((size_t)(ti0 * 16) + rr) * C_DIM;
    const float* xcol = X + ((size_t)bb * C_DIM) * HW + hw;

    v8f acc[4] = {};
    for (int k = 0; k < C_DIM; k += 4) {
        const int k0 = k + g * 2;
        v2f b;
        b.x = xcol[(size_t)k0 * HW] - mean[k0];
        b.y = xcol[(size_t)(k0 + 1) * HW] - mean[k0 + 1];
#pragma unroll
        for (int t = 0; t < 4; ++t) {
            v2f a;
            a.x = arow[t * 16 * C_DIM + k0];       // contiguous pair -> b64
            a.y = arow[t * 16 * C_DIM + k0 + 1];
            acc[t] = WMMA_F32(a, b, acc[t]);
        }
    }

    const float s = scalars[1];         // sqrt(rTr)
#pragma unroll
    for (int t = 0; t < 4; ++t) {
        const int rowbase = (ti0 + t) * 16 + g * 8;
#pragma unroll
        for (int r = 0; r < 8; ++r)
            out[((size_t)(bb * C_DIM + rowbase + r)) * HW + hw] = acc[t][r] * s;
    }
}

// ---------------------------------------------------------------------------
extern "C" void kernel_launch(void* const* d_in, const int* in_sizes, int n_in,
                              void* d_out, int out_size, void* d_ws, size_t ws_size,
                              hipStream_t stream) {
    (void)in_sizes; (void)n_in; (void)out_size; (void)ws_size;
    const float* X = (const float*)d_in[0];
    float* out = (float*)d_out;
    float* ws = (float*)d_ws;

    float* mean    = ws;                  // 256
    float* scalars = ws + 256;            // 2 (rTr, sqrt(rTr))
    float* Sigma   = ws + 512;            // 65536
    float* SigmaN  = Sigma  + C_DIM * C_DIM;
    float* P       = SigmaN + C_DIM * C_DIM;
    float* T1      = P      + C_DIM * C_DIM;
    float* T2      = T1     + C_DIM * C_DIM;

    mean_kernel<<<C_DIM, 256, 0, stream>>>(X, mean);
    sigma_init_kernel<<<(C_DIM * C_DIM + 255) / 256, 256, 0, stream>>>(Sigma);
    sigma_kernel<<<dim3(8, 8, B_DIM), 32, 0, stream>>>(X, mean, Sigma);
    trace_kernel<<<1, 256, 0, stream>>>(Sigma, scalars);
    scale_init_kernel<<<(C_DIM * C_DIM + 255) / 256, 256, 0, stream>>>(Sigma, scalars, SigmaN, P);

    for (int t = 0; t < T_ITERS; ++t) {
        matmul256_kernel<<<dim3(16, 16), 32, 0, stream>>>(P, P, T1);     // T1 = P^2
        matmul256_kernel<<<dim3(16, 16), 32, 0, stream>>>(T1, P, T2);    // T2 = P^3
        newton_update_kernel<<<dim3(16, 16), 32, 0, stream>>>(T2, SigmaN, P);
    }

    whiten_kernel<<<dim3(4, M_TOT / 16), 32, 0, stream>>>(X, mean, P, scalars, out);
}